// TripletLoss_76579266888382
// MI455X (gfx1250) — compile-verified
//
#include <hip/hip_runtime.h>
#include <hip/hip_bf16.h>
#include <math.h>

typedef __attribute__((ext_vector_type(2))) float v2f;
typedef __attribute__((ext_vector_type(4))) float v4f;
typedef __attribute__((ext_vector_type(8))) float v8f;

#define NROWS 4096
#define KDIM  2048
#define MARGIN_F 0.3f
#define SHIFT_AP 96.0f

static __device__ __forceinline__ v8f wmma_f32_4(v2f a, v2f b, v8f c) {
    // V_WMMA_F32_16X16X4_F32: D = A(16x4) * B(4x16) + C, full f32 accumulate
    return __builtin_amdgcn_wmma_f32_16x16x4_f32(
        /*neg_a=*/false, a, /*neg_b=*/false, b,
        /*c_mod=*/(short)0, c, /*reuse_a=*/false, /*reuse_b=*/false);
}

__global__ void zero_out_kernel(float* out) {
    if (threadIdx.x == 0) out[0] = 0.0f;
}

// One block per row: sq[row] = sum_k X[row][k]^2  (exact f32, matches reference)
__global__ __launch_bounds__(256) void rowsq_kernel(const float* __restrict__ X,
                                                    float* __restrict__ sq) {
    const int row = blockIdx.x;
    const float* xr = X + (size_t)row * KDIM;
    float acc = 0.0f;
    for (int k = threadIdx.x; k < KDIM; k += 256) {
        float v = xr[k];
        acc = fmaf(v, v, acc);
    }
    __shared__ float red[256];
    red[threadIdx.x] = acc;
    __syncthreads();
    for (int s = 128; s > 0; s >>= 1) {
        if (threadIdx.x < s) red[threadIdx.x] += red[threadIdx.x + s];
        __syncthreads();
    }
    if (threadIdx.x == 0) sq[row] = red[0];
}

// Fused Gram(WMMA f32) -> distance -> shifted-softmax accumulation -> loss.
// Block = 256 threads = 8 waves; block owns 16 i-rows.
// Each wave computes a 16x64 strip per j-tile (4 C tiles reuse one A load):
// per 8-K group: 1 A b128 + 4 B b128 loads feed 8 WMMAs.
__global__ __launch_bounds__(256) void triplet_fused_kernel(const float* __restrict__ X,
                                                            const int*  __restrict__ tgt,
                                                            const float* __restrict__ sq,
                                                            float* __restrict__ out) {
    const int lane = threadIdx.x & 31;
    const int wave = threadIdx.x >> 5;     // 0..7
    const int lm   = lane & 15;
    const int hi   = lane >> 4;            // lane half selects K sub-slot
    const int i0   = blockIdx.x * 16;      // 16 rows per block

    __shared__ float sh_sap[16], sh_wap[16], sh_san[16], sh_wan[16];
    __shared__ float sh_sqr[16];
    __shared__ int   sh_tgt[16];

    if (threadIdx.x < 16) {
        sh_sap[threadIdx.x] = 0.0f; sh_wap[threadIdx.x] = 0.0f;
        sh_san[threadIdx.x] = 0.0f; sh_wan[threadIdx.x] = 0.0f;
        sh_sqr[threadIdx.x] = sq[i0 + threadIdx.x];
        sh_tgt[threadIdx.x] = tgt[i0 + threadIdx.x];
    }
    __syncthreads();

    // Per-lane softmax partials for row-slots r=0..7 (global row = i0 + 8*hi + r)
    float acc_sap[8], acc_wap[8], acc_san[8], acc_wan[8];
#pragma unroll
    for (int r = 0; r < 8; ++r) { acc_sap[r] = 0.f; acc_wap[r] = 0.f; acc_san[r] = 0.f; acc_wan[r] = 0.f; }

    // A rows fixed for the whole block: row i0+lm, both lane halves (hi picks K slot)
    const float* Abase = X + (size_t)(i0 + lm) * KDIM;

    for (int jt = wave; jt < NROWS / 64; jt += 8) {
        const int j0 = jt * 64;

        const float* Bbase[4];
#pragma unroll
        for (int t = 0; t < 4; ++t)
            Bbase[t] = X + (size_t)(j0 + 16 * t + lm) * KDIM;   // cols j0+16t .. j0+16t+15

        v8f c[4] = { {}, {}, {}, {} };

        // K-loop: b128 loads cover 8 K each (permutation-invariant dot product:
        // lane half hi consumes K = k0+4*hi .. k0+4*hi+3; two WMMAs use .xy / .zw
        // slots consistently for A and B, so every k pairs correctly exactly once).
#pragma unroll 2
        for (int k0 = 0; k0 < KDIM; k0 += 8) {
            const int off = k0 + 4 * hi;
            v4f a = *(const v4f*)(Abase + off);
            v4f b[4];
#pragma unroll
            for (int t = 0; t < 4; ++t)
                b[t] = *(const v4f*)(Bbase[t] + off);
#pragma unroll
            for (int t = 0; t < 4; ++t) {
                c[t] = wmma_f32_4(a.xy, b[t].xy, c[t]);
                c[t] = wmma_f32_4(a.zw, b[t].zw, c[t]);
            }
        }

        // Epilogue: Gram -> distance -> shifted softmax partial sums (4 cols/lane).
        float sqn[4];
        int   tn[4];
#pragma unroll
        for (int t = 0; t < 4; ++t) {
            sqn[t] = sq[j0 + 16 * t + lm];
            tn[t]  = tgt[j0 + 16 * t + lm];
        }
#pragma unroll
        for (int r = 0; r < 8; ++r) {
            const int slot = 8 * hi + r;          // row-in-block, per C/D VGPR layout
            const float sm = sh_sqr[slot];
            const int   tm = sh_tgt[slot];
#pragma unroll
            for (int t = 0; t < 4; ++t) {
                float d2 = sm + sqn[t] - 2.0f * c[t][r];
                float d  = sqrtf(fmaxf(d2, 1e-12f));
                if (tm == tn[t]) { float e = __expf(d - SHIFT_AP); acc_sap[r] += e; acc_wap[r] += d * e; }
                else             { float e = __expf(-d);           acc_san[r] += e; acc_wan[r] += d * e; }
            }
        }
    }

    // Merge per-lane partials across the block (ds_add_f32).
#pragma unroll
    for (int r = 0; r < 8; ++r) {
        const int slot = 8 * hi + r;
        atomicAdd(&sh_sap[slot], acc_sap[r]);
        atomicAdd(&sh_wap[slot], acc_wap[r]);
        atomicAdd(&sh_san[slot], acc_san[r]);
        atomicAdd(&sh_wan[slot], acc_wan[r]);
    }
    __syncthreads();

    if (threadIdx.x < 16) {
        const float dap = sh_wap[threadIdx.x] / sh_sap[threadIdx.x];
        const float dan = sh_wan[threadIdx.x] / sh_san[threadIdx.x];
        const float l = fmaxf(dap - dan + MARGIN_F, 0.0f) * (1.0f / (float)NROWS);
        atomicAdd(out, l);  // global_atomic_add_f32
    }
}

extern "C" void kernel_launch(void* const* d_in, const int* in_sizes, int n_in,
                              void* d_out, int out_size, void* d_ws, size_t ws_size,
                              hipStream_t stream) {
    (void)in_sizes; (void)n_in; (void)out_size; (void)ws_size;
    const float* X   = (const float*)d_in[0];
    const int*   tgt = (const int*)d_in[1];
    float*       out = (float*)d_out;
    float*       sq  = (float*)d_ws;  // 4096 floats = 16 KB scratch

    zero_out_kernel<<<1, 64, 0, stream>>>(out);
    rowsq_kernel<<<NROWS, 256, 0, stream>>>(X, sq);
    triplet_fused_kernel<<<NROWS / 16, 256, 0, stream>>>(X, tgt, sq, out);
}